// Decoder_86758339379419
// MI455X (gfx1250) — compile-verified
//
#include <hip/hip_runtime.h>
#include <stdint.h>

#define NBOX            98
#define PRED_PER_BATCH  1470      // 49 cells * 30 floats
#define SORT_N          128
#define OBJ_THR         0.1f
#define NMS_THR         0.5f

__global__ __launch_bounds__(128) void yolo_decode_nms_kernel(
    const float* __restrict__ pred, float* __restrict__ out, int B)
{
    __shared__ float    s_pred[PRED_PER_BATCH];
    __shared__ float    s_dec[6][NBOX];       // x1,y1,x2,y2,cls,score (unsorted)
    __shared__ uint64_t s_skey[SORT_N];       // packed sort key: score bits | (127 - idx)
    __shared__ int      s_idx[NBOX];          // sorted position -> original box
    __shared__ float4   s_sbox[NBOX];         // sorted x1,y1,x2,y2
    __shared__ float    s_area[NBOX];
    __shared__ uint4    s_adj[NBOX];          // upper-tri suppression bitmask rows
    __shared__ uint32_t s_amask[4];           // alive (valid) bitmask
    __shared__ int      s_keep[NBOX];

    const int tid = threadIdx.x;
    const int bb  = blockIdx.x;

    // ---------- Stage this batch's 1470 floats (5880 B) global -> LDS via
    // gfx1250 async DMA (ASYNCcnt path). 735 chunks of 8 bytes, GVS mode. ----------
    {
        const uint64_t gbase = (uint64_t)(uintptr_t)(pred + (size_t)bb * PRED_PER_BATCH);
        const uint32_t lbase = (uint32_t)(uintptr_t)(void*)s_pred;
        #pragma unroll
        for (int r = 0; r < 6; ++r) {
            int c = tid + r * 128;
            if (c < 735) {
                uint32_t ldst = lbase + (uint32_t)c * 8u;
                uint32_t goff = (uint32_t)c * 8u;
                asm volatile("global_load_async_to_lds_b64 %0, %1, %2"
                             :: "v"(ldst), "v"(goff), "s"(gbase)
                             : "memory");
            }
        }
        asm volatile("s_wait_asynccnt 0" ::: "memory");
    }
    __syncthreads();

    // ---------- Decode: one thread per box ----------
    if (tid < 4) s_amask[tid] = 0;            // zeroed before any atomicOr (sort barriers separate)
    if (tid < NBOX) {
        const int cell  = tid >> 1;
        const int which = tid & 1;
        const int row   = cell / 7;           // gy axis (axis 1)
        const int col   = cell % 7;           // gx axis (axis 2)
        const float* cp = &s_pred[cell * 30];

        float contain = cp[which ? 9 : 4];
        float maxc = cp[10];
        int   maxi = 0;
        #pragma unroll
        for (int c = 1; c < 20; ++c) {
            float v = cp[10 + c];
            if (v > maxc) { maxc = v; maxi = c; }   // strict > == first-max (jnp.argmax)
        }
        float prob = contain * maxc;
        float mm   = (prob > OBJ_THR) ? 1.0f : 0.0f;

        const int bo = which * 5;             // fields 0..3 or 5..8
        float bx = cp[bo + 0], by = cp[bo + 1], bw = cp[bo + 2], bh = cp[bo + 3];
        float cx = (bx + (float)col) / 7.0f;
        float cy = (by + (float)row) / 7.0f;
        float x1 = fmaxf(cx - 0.5f * bw, 0.0f) * mm;
        float y1 = fmaxf(cy - 0.5f * bh, 0.0f) * mm;
        float x2 = fminf(cx + 0.5f * bw, 1.0f) * mm;
        float y2 = fminf(cy + 0.5f * bh, 1.0f) * mm;

        s_dec[0][tid] = x1;
        s_dec[1][tid] = y1;
        s_dec[2][tid] = x2;
        s_dec[3][tid] = y2;
        s_dec[4][tid] = (float)maxi * mm;
        s_dec[5][tid] = prob * mm;

        float score = prob * mm;              // >= 0, never NaN
        s_skey[tid] = ((uint64_t)__float_as_uint(score) << 32) |
                      (uint32_t)((SORT_N - 1) - tid);
        s_keep[tid] = 0;
    } else {
        // pad: score bits = 0; idx tie-break puts pads after all real boxes
        s_skey[tid] = (uint32_t)((SORT_N - 1) - tid);
    }

    // ---------- Bitonic sort of 128 packed u64 keys, descending.
    // Wave-local sync (compiler fence only) when both this step and the
    // previous step exchange within a wave32 (partner = tid^j, j < 32). ----------
    for (int k = 2; k <= SORT_N; k <<= 1) {
        for (int j = k >> 1; j > 0; j >>= 1) {
            const bool full = (j >= 32) || ((2 * j >= 32) && (2 * j <= (k >> 1)));
            if (full) {
                __syncthreads();
            } else {
                __builtin_amdgcn_wave_barrier();
                asm volatile("" ::: "memory");
            }
            int ixj = tid ^ j;
            if (ixj > tid) {
                uint64_t a = s_skey[tid];
                uint64_t b = s_skey[ixj];
                bool aAfterB = a < b;                         // descending order
                bool doSwap  = ((tid & k) == 0) ? aAfterB : !aAfterB;
                if (doSwap) { s_skey[tid] = b; s_skey[ixj] = a; }
            }
        }
    }
    __syncthreads();

    // ---------- Gather sorted boxes; build valid-box bitmask via ds_or ----------
    if (tid < NBOX) {
        int o = (SORT_N - 1) - (int)(uint32_t)s_skey[tid];
        s_idx[tid] = o;
        float4 b = make_float4(s_dec[0][o], s_dec[1][o], s_dec[2][o], s_dec[3][o]);
        s_sbox[tid] = b;
        s_area[tid] = (b.z - b.x) * (b.w - b.y);
        if (s_dec[5][o] > 0.0f)
            atomicOr(&s_amask[tid >> 5], 1u << (tid & 31));
    }
    __syncthreads();

    // ---------- Parallel upper-triangular suppression adjacency.
    // Row i, bit j (j>i) set iff IoU(i,j) > thr; divide-free compare. ----------
    if (tid < NBOX) {
        const int i  = tid;
        float4 bi    = s_sbox[i];
        float  ai    = s_area[i];
        uint32_t* rowp = (uint32_t*)&s_adj[i];
        #pragma unroll
        for (int w = 0; w < 4; ++w) {
            uint32_t m = 0;
            const int j0  = w * 32;
            const int jhi = (w < 3) ? 32 : (NBOX - 96);
            int jj = (i + 1 > j0) ? (i + 1 - j0) : 0;
            #pragma unroll 2
            for (; jj < jhi; ++jj) {
                const int j = j0 + jj;
                float4 bj = s_sbox[j];
                float ix1 = fmaxf(bi.x, bj.x);
                float iy1 = fmaxf(bi.y, bj.y);
                float ix2 = fminf(bi.z, bj.z);
                float iy2 = fminf(bi.w, bj.w);
                float inter = fmaxf(ix2 - ix1, 0.0f) * fmaxf(iy2 - iy1, 0.0f);
                float uni   = ai + s_area[j] - inter;
                bool  sup   = (uni > 0.0f) && (inter > NMS_THR * uni);   // iou > thr
                m |= (sup ? 1u : 0u) << jj;
            }
            rowp[w] = m;
        }
    }
    __syncthreads();

    // ---------- Greedy scan: visit only kept boxes via find-first-set ----------
    if (tid == 0) {
        uint32_t a0 = s_amask[0], a1 = s_amask[1], a2 = s_amask[2], a3 = s_amask[3];
        #define SCAN_WORD(W, AW)                                           \
        {                                                                  \
            uint32_t aw = (AW);                                            \
            while (aw) {                                                   \
                int b = __builtin_ctz(aw);                                 \
                uint4 r = s_adj[(W) * 32 + b];   /* ds_load_b128 */        \
                s_keep[(W) * 32 + b] = 1;                                  \
                a0 &= ~r.x; a1 &= ~r.y; a2 &= ~r.z; a3 &= ~r.w;            \
                aw = (AW) & ~((2u << b) - 1u);   /* bits strictly above b */\
            }                                                              \
        }
        SCAN_WORD(0, a0)
        SCAN_WORD(1, a1)
        SCAN_WORD(2, a2)
        SCAN_WORD(3, a3)
        #undef SCAN_WORD
    }
    __syncthreads();

    // ---------- Coalesced outputs ----------
    const size_t ob = (size_t)bb * (NBOX * 6);
    for (int e = tid; e < NBOX * 6; e += 128) {
        int p = e / 6, f = e % 6;
        float kf = s_keep[p] ? 1.0f : 0.0f;
        out[ob + e] = s_dec[f][s_idx[p]] * kf;
    }
    float* keep_out = out + (size_t)B * (NBOX * 6);
    if (tid < NBOX)
        keep_out[(size_t)bb * NBOX + tid] = s_keep[tid] ? 1.0f : 0.0f;
}

extern "C" void kernel_launch(void* const* d_in, const int* in_sizes, int n_in,
                              void* d_out, int out_size, void* d_ws, size_t ws_size,
                              hipStream_t stream)
{
    const float* pred = (const float*)d_in[0];
    float*       out  = (float*)d_out;
    const int B = in_sizes[0] / PRED_PER_BATCH;   // 16384
    yolo_decode_nms_kernel<<<dim3(B), dim3(128), 0, stream>>>(pred, out, B);
}